// GumbelSinkhornModel_53283364274809
// MI455X (gfx1250) — compile-verified
//
#include <hip/hip_runtime.h>
#include <cstdint>
#include <cstddef>

// Gumbel-Sinkhorn forward, fused for MI455X (gfx1250, wave32).
// grid = 128 (one 512x512 matrix per workgroup), block = 1024 (32 waves).
// Element->lane map: col = 4*(m*32+lane)+c, m=0..3, c=0..3 (float4 per m).
// Row r = wave*16 + k, k=0..15 (each wave owns 16 rows for the whole kernel).
//
// Cache strategy (ISA ch.7 TH tables):
//  - P stores use th:TH_STORE_WB -> force-dirty in L2, override default
//    wr-rinse so the 134MB iterating tensor never rinses to HBM (fits in
//    192MB L2; ~19GB of stores over 150 iterations stay on-chip).
//  - logits/u1 are read exactly once -> TH_LOAD_NT so they don't evict P.

#define GS_N    512
#define GS_IT   150
#define GS_EPSF 1e-20f

typedef __attribute__((ext_vector_type(4))) float v4f;

__device__ __forceinline__ float wave_sum32(float v) {
#pragma unroll
  for (int o = 16; o > 0; o >>= 1) v += __shfl_xor(v, o, 32);
  return v;
}
__device__ __forceinline__ float wave_max32(float v) {
#pragma unroll
  for (int o = 16; o > 0; o >>= 1) v = fmaxf(v, __shfl_xor(v, o, 32));
  return v;
}

__device__ __forceinline__ float f4_max(const float4 a) {
  return fmaxf(fmaxf(a.x, a.y), fmaxf(a.z, a.w));
}
__device__ __forceinline__ float f4_sum(const float4 a) {
  return (a.x + a.y) + (a.z + a.w);
}

// Non-temporal 16B load (th:TH_LOAD_NT) via clang ext-vector type.
__device__ __forceinline__ float4 load_nt_b128(const float* gsrc) {
  const v4f t = __builtin_nontemporal_load((const v4f*)gsrc);
  return make_float4(t.x, t.y, t.z, t.w);
}

// 16B store with WB temporal hint: keep line dirty in far cache (L2), no rinse.
__device__ __forceinline__ void store_wb_b128(float* gdst, const float4 w) {
  v4f d;
  d.x = w.x; d.y = w.y; d.z = w.z; d.w = w.w;
  const uint64_t ga = (uint64_t)(uintptr_t)gdst;
  asm volatile("global_store_b128 %0, %1, off th:TH_STORE_WB" :: "v"(ga), "v"(d) : "memory");
}

// Async copy one 2KB row (512 f32) global -> LDS across the 32 lanes:
// 4 x global_load_async_to_lds_b128 (16B per lane per op). Tracked by ASYNCcnt.
__device__ __forceinline__ void async_row(const float* gsrc, uint32_t lds_byte, int lane) {
#pragma unroll
  for (int m = 0; m < 4; ++m) {
    uint64_t ga = (uint64_t)(uintptr_t)gsrc + (uint64_t)(((unsigned)(m * 32 + lane)) << 4);
    uint32_t la = lds_byte + (((unsigned)(m * 32 + lane)) << 4);
    asm volatile("global_load_async_to_lds_b128 %0, %1, off" :: "v"(la), "v"(ga) : "memory");
  }
}

// Flush per-lane column partials -> LDS colsum (ds_add_f32), build cinv for the
// next iteration, re-zero colsum, and reload this lane's 16 cinv values.
__device__ __forceinline__ void fold_columns(float* colsum, float* cinvs,
                                             float4 (&cacc)[4], float4 (&cin)[4],
                                             int tid, int lane) {
#pragma unroll
  for (int m = 0; m < 4; ++m) {
    const int cb = (m * 32 + lane) * 4;
    atomicAdd(&colsum[cb + 0], cacc[m].x);
    atomicAdd(&colsum[cb + 1], cacc[m].y);
    atomicAdd(&colsum[cb + 2], cacc[m].z);
    atomicAdd(&colsum[cb + 3], cacc[m].w);
  }
  __syncthreads();
  if (tid < GS_N) {
    const float c = colsum[tid];
    cinvs[tid] = (c > GS_EPSF) ? (1.0f / c) : 1.0f;
    colsum[tid] = 0.0f;
  }
  __syncthreads();
#pragma unroll
  for (int m = 0; m < 4; ++m) {
    cin[m] = *(const float4*)(cinvs + (m * 32 + lane) * 4);
    cacc[m] = make_float4(0.f, 0.f, 0.f, 0.f);
  }
}

__global__ void __launch_bounds__(1024, 1)
gumbel_sinkhorn_fused(const float* __restrict__ logits,
                      const float* __restrict__ u1,
                      const int* __restrict__ steps_p,
                      float* __restrict__ out) {
  extern __shared__ float smem[];
  float* colsum = smem;             // [512] column-sum accumulator
  float* cinvs  = smem + GS_N;      // [512] 1/colsum for current iteration
  float* rinvs  = smem + 2 * GS_N;  // [512] 1/rowsum (each row owned by one wave)
  float* rowbuf = smem + 3 * GS_N;  // [32 waves][2 bufs][512] async staging

  const int tid  = (int)threadIdx.x;
  const int lane = tid & 31;
  const int wv   = tid >> 5;
  const size_t mbase = (size_t)blockIdx.x * GS_N * GS_N;

  if (tid < GS_N) colsum[tid] = 0.0f;
  __syncthreads();

  const float tau  = fmaxf(powf(0.9999f, (float)steps_p[0]), 0.1f);
  const float itau = 1.0f / tau;

  float4 cacc[4], cin[4];
#pragma unroll
  for (int m = 0; m < 4; ++m) cacc[m] = make_float4(0.f, 0.f, 0.f, 0.f);

  // ---------------- Phase 1: P0 + rowsum + (row-normalized) colsum ----------
  // P0 = softmax(logits + 0.015 * softmax((logits + gumbel(u1)) / tau))
  // (the u2 straight-through term is exactly zero in the forward value)
#pragma unroll 1
  for (int k = 0; k < 16; ++k) {
    const int row = wv * 16 + k;
    const size_t rb = mbase + (size_t)row * GS_N;

    float4 lg[4], tv[4];
    float mx = -3.402823466e38f;
#pragma unroll
    for (int m = 0; m < 4; ++m) {
      lg[m] = load_nt_b128(logits + rb + (size_t)((m * 32 + lane) * 4));  // read-once
      const float4 u = load_nt_b128(u1 + rb + (size_t)((m * 32 + lane) * 4));
      float4 t;
      t.x = (lg[m].x - logf(-logf(u.x + GS_EPSF) + GS_EPSF)) * itau;
      t.y = (lg[m].y - logf(-logf(u.y + GS_EPSF) + GS_EPSF)) * itau;
      t.z = (lg[m].z - logf(-logf(u.z + GS_EPSF) + GS_EPSF)) * itau;
      t.w = (lg[m].w - logf(-logf(u.w + GS_EPSF) + GS_EPSF)) * itau;
      tv[m] = t;
      mx = fmaxf(mx, f4_max(t));
    }
    mx = wave_max32(mx);

    float s1 = 0.f;
#pragma unroll
    for (int m = 0; m < 4; ++m) {
      tv[m].x = expf(tv[m].x - mx);
      tv[m].y = expf(tv[m].y - mx);
      tv[m].z = expf(tv[m].z - mx);
      tv[m].w = expf(tv[m].w - mx);
      s1 += f4_sum(tv[m]);
    }
    s1 = wave_sum32(s1);
    const float a = 0.015f / s1;  // NOISE_FACTOR * (1/sum)

    float mx2 = -3.402823466e38f;
#pragma unroll
    for (int m = 0; m < 4; ++m) {
      lg[m].x = lg[m].x + a * tv[m].x;
      lg[m].y = lg[m].y + a * tv[m].y;
      lg[m].z = lg[m].z + a * tv[m].z;
      lg[m].w = lg[m].w + a * tv[m].w;
      mx2 = fmaxf(mx2, f4_max(lg[m]));
    }
    mx2 = wave_max32(mx2);

    float s2 = 0.f;
#pragma unroll
    for (int m = 0; m < 4; ++m) {
      lg[m].x = expf(lg[m].x - mx2);
      lg[m].y = expf(lg[m].y - mx2);
      lg[m].z = expf(lg[m].z - mx2);
      lg[m].w = expf(lg[m].w - mx2);
      s2 += f4_sum(lg[m]);
    }
    s2 = wave_sum32(s2);
    const float is2 = 1.0f / s2;

    float rs = 0.f;
#pragma unroll
    for (int m = 0; m < 4; ++m) {
      lg[m].x *= is2; lg[m].y *= is2; lg[m].z *= is2; lg[m].w *= is2;
      rs += f4_sum(lg[m]);
      store_wb_b128(out + rb + (size_t)((m * 32 + lane) * 4), lg[m]);
    }
    rs = wave_sum32(rs);
    const float rn = (rs > GS_EPSF) ? (1.0f / rs) : 1.0f;
    if (lane == 0) rinvs[row] = rn;
#pragma unroll
    for (int m = 0; m < 4; ++m) {
      cacc[m].x += lg[m].x * rn;
      cacc[m].y += lg[m].y * rn;
      cacc[m].z += lg[m].z * rn;
      cacc[m].w += lg[m].w * rn;
    }
  }
  fold_columns(colsum, cinvs, cacc, cin, tid, lane);

  const uint32_t ldsrow0 = (uint32_t)(uintptr_t)(rowbuf + wv * 1024);  // LDS byte offset (low 32 bits of generic ptr)

  // ---------------- 150 fused Sinkhorn iterations ---------------------------
  // One pass/iteration: read P row via async->LDS (double buffered),
  // P = clip(P * rinv * cinv), WB-store, wave-reduce new rowsum -> rinv(next),
  // accumulate v*rinv(next) into column partials -> cinv(next).
#pragma unroll 1
  for (int it = 0; it < GS_IT; ++it) {
    // Order in-place stores (STOREcnt) vs async loads (ASYNCcnt) on same rows.
    asm volatile("s_wait_storecnt 0x0" ::: "memory");
    async_row(out + mbase + (size_t)(wv * 16) * GS_N, ldsrow0, lane);

#pragma unroll 1
    for (int k = 0; k < 16; ++k) {
      const int row = wv * 16 + k;
      if (k < 15) {
        async_row(out + mbase + (size_t)(row + 1) * GS_N,
                  ldsrow0 + (uint32_t)(((k + 1) & 1) * 2048), lane);
        asm volatile("s_wait_asynccnt 0x4" ::: "memory");  // row k done, k+1 in flight
      } else {
        asm volatile("s_wait_asynccnt 0x0" ::: "memory");
      }

      const float rc = rinvs[row];  // wave-uniform LDS broadcast
      const float4* pb = (const float4*)(rowbuf + wv * 1024 + (k & 1) * 512);

      float4 v[4];
      float rs = 0.f;
#pragma unroll
      for (int m = 0; m < 4; ++m) {
        const float4 p = pb[m * 32 + lane];
        float4 w;
        w.x = fminf(fmaxf(p.x * rc * cin[m].x, GS_EPSF), 1.0f);
        w.y = fminf(fmaxf(p.y * rc * cin[m].y, GS_EPSF), 1.0f);
        w.z = fminf(fmaxf(p.z * rc * cin[m].z, GS_EPSF), 1.0f);
        w.w = fminf(fmaxf(p.w * rc * cin[m].w, GS_EPSF), 1.0f);
        v[m] = w;
        rs += f4_sum(w);
      }

      float* orow = out + mbase + (size_t)row * GS_N;
#pragma unroll
      for (int m = 0; m < 4; ++m)
        store_wb_b128(orow + (size_t)((m * 32 + lane) * 4), v[m]);

      rs = wave_sum32(rs);
      const float rn = (rs > GS_EPSF) ? (1.0f / rs) : 1.0f;
      if (lane == 0) rinvs[row] = rn;
#pragma unroll
      for (int m = 0; m < 4; ++m) {
        cacc[m].x += v[m].x * rn;
        cacc[m].y += v[m].y * rn;
        cacc[m].z += v[m].z * rn;
        cacc[m].w += v[m].w * rn;
      }
    }
    fold_columns(colsum, cinvs, cacc, cin, tid, lane);
  }
}

extern "C" void kernel_launch(void* const* d_in, const int* in_sizes, int n_in,
                              void* d_out, int out_size, void* d_ws, size_t ws_size,
                              hipStream_t stream) {
  (void)in_sizes; (void)n_in; (void)d_ws; (void)ws_size; (void)out_size;
  const float* logits = (const float*)d_in[0];
  const float* u1     = (const float*)d_in[1];
  // d_in[2] (u2) is unused in the forward value: st - stop_gradient(st) == 0.
  const int* steps    = (const int*)d_in[3];
  float* out          = (float*)d_out;

  const size_t shmem = (size_t)(3 * GS_N + 32 * 2 * GS_N) * sizeof(float);  // 137216 B < 320KB/WGP
  gumbel_sinkhorn_fused<<<dim3(128), dim3(1024), shmem, stream>>>(logits, u1, steps, out);
}